// GHMC_52853867545035
// MI455X (gfx1250) — compile-verified
//
#include <hip/hip_runtime.h>
#include <stdint.h>

#define BINS 10
#define TPB  256
#define NBLK 2048

typedef __attribute__((ext_vector_type(2))) float v2f;
typedef __attribute__((ext_vector_type(8))) float v8f;

// ---------------------------------------------------------------------------
// CDNA5 async global->LDS copy (ASYNCcnt-tracked), per CDNA5 ISA
// Each active lane copies 16B from its global address to its LDS offset.
// LDS offset = low 32 bits of the flat pointer (ISA 10.2 LDS aperture rule).
// ---------------------------------------------------------------------------
__device__ __forceinline__ void async_copy_b128(void* lds_dst, const void* gsrc) {
    unsigned lds_off = (unsigned)(uintptr_t)lds_dst;
    asm volatile("global_load_async_to_lds_b128 %0, %1, off"
                 :: "v"(lds_off), "v"(gsrc)
                 : "memory");
}
__device__ __forceinline__ void wait_async_le2() {
    asm volatile("s_wait_asynccnt 2" ::: "memory");
}
__device__ __forceinline__ void wait_async_le0() {
    asm volatile("s_wait_asynccnt 0" ::: "memory");
}

__device__ __forceinline__ float frcp(float x) { return __builtin_amdgcn_rcpf(x); }

// ---------------------------------------------------------------------------
// Exact fp32 wave(32) sum via V_WMMA_F32_16X16X4_F32 ones-matrix adder trees.
// Deterministic (fixed hardware accumulation order), full fp32 precision.
// Requires EXEC == all ones (call only from fully-converged code).
//   Step 1: A = per-lane {acc,0} (16x4), B = ones  -> D[m][n] = acc[m]+acc[m+16]
//   Step 2: A = ones, B = VGPR pairs of step-1 D, chained via C accumulator
// Every lane's d[0] ends up holding the full 32-lane sum.
// ---------------------------------------------------------------------------
__device__ __forceinline__ float wave_sum_wmma(float acc) {
    v2f ones; ones.x = 1.0f; ones.y = 1.0f;
    v2f a;    a.x = acc;     a.y = 0.0f;
    v8f zero = {};
    v8f s = __builtin_amdgcn_wmma_f32_16x16x4_f32(false, a, false, ones,
                                                  (short)0, zero, false, false);
    v8f d = {};
    v2f b;
    b.x = s[0]; b.y = s[1];
    d = __builtin_amdgcn_wmma_f32_16x16x4_f32(false, ones, false, b, (short)0, d, false, false);
    b.x = s[2]; b.y = s[3];
    d = __builtin_amdgcn_wmma_f32_16x16x4_f32(false, ones, false, b, (short)0, d, false, false);
    b.x = s[4]; b.y = s[5];
    d = __builtin_amdgcn_wmma_f32_16x16x4_f32(false, ones, false, b, (short)0, d, false, false);
    b.x = s[6]; b.y = s[7];
    d = __builtin_amdgcn_wmma_f32_16x16x4_f32(false, ones, false, b, (short)0, d, false, false);
    return d[0];
}

// g = |sigmoid(x) - t| ; bin = clip(floor(10g), 0, 9). 1 exp + 1 rcp.
__device__ __forceinline__ int bin_of(float x, float t) {
    float e   = __expf(-fabsf(x));        // v_exp_f32
    float r   = frcp(1.0f + e);           // v_rcp_f32
    float sig = (x >= 0.0f) ? r : e * r;  // sigmoid(x)
    float g   = fabsf(sig - t);           // in [0,1]
    int   b   = (int)(g * 10.0f);         // g>=0 -> trunc == floor
    return b > (BINS - 1) ? (BINS - 1) : b;
}

// bin + numerically-stable BCE-with-logits. 1 exp + 1 rcp + 1 log.
__device__ __forceinline__ void elem_of(float x, float t, int& idx, float& bce) {
    float e   = __expf(-fabsf(x));
    float r   = frcp(1.0f + e);
    float sig = (x >= 0.0f) ? r : e * r;
    float g   = fabsf(sig - t);
    int   b   = (int)(g * 10.0f);
    idx = b > (BINS - 1) ? (BINS - 1) : b;
    bce = fmaxf(x, 0.0f) - x * t + __logf(1.0f + e);  // log1p(exp(-|x|))
}

// ---------------------------------------------------------------------------
// Kernel 0: zero the global histogram (ws is poisoned, not re-zeroed by harness)
// ---------------------------------------------------------------------------
__global__ void k_zero(unsigned* __restrict__ counts) {
    if (threadIdx.x < BINS) counts[threadIdx.x] = 0u;
}

// ---------------------------------------------------------------------------
// Kernel 1: global 10-bin histogram. Integer-only accumulation => deterministic.
// Byte-packed per-thread counters (~9 VALU/elem vs ~20 for 10x cmp-add),
// flushed every 63 float4 iterations (worst case 252 <= 255 per byte).
// ---------------------------------------------------------------------------
__global__ void k_hist(const float* __restrict__ pred, const float* __restrict__ tgt,
                       unsigned* __restrict__ gcounts, long nvec, int rem) {
    __shared__ unsigned hist[BINS];
    if (threadIdx.x < BINS) hist[threadIdx.x] = 0u;
    __syncthreads();

    unsigned cnt[BINS];
#pragma unroll
    for (int b = 0; b < BINS; ++b) cnt[b] = 0u;

    unsigned c0 = 0u, c1 = 0u, c2 = 0u;   // bins 0-3 / 4-7 / 8-9, 8 bits each

    const float4* p4 = (const float4*)pred;
    const float4* t4 = (const float4*)tgt;
    long gid    = (long)blockIdx.x * blockDim.x + threadIdx.x;
    long stride = (long)gridDim.x * blockDim.x;

    long v = gid;
    while (v < nvec) {
        int burst = 0;
        for (; v < nvec && burst < 63; v += stride, ++burst) {
            float4 x = p4[v];            // global_load_b128
            float4 t = t4[v];
            int b0 = bin_of(x.x, t.x);
            int b1 = bin_of(x.y, t.y);
            int b2 = bin_of(x.z, t.z);
            int b3 = bin_of(x.w, t.w);
            unsigned i0 = 1u << ((b0 & 3) << 3), s0 = (unsigned)(b0 >> 2);
            unsigned i1 = 1u << ((b1 & 3) << 3), s1 = (unsigned)(b1 >> 2);
            unsigned i2 = 1u << ((b2 & 3) << 3), s2 = (unsigned)(b2 >> 2);
            unsigned i3 = 1u << ((b3 & 3) << 3), s3 = (unsigned)(b3 >> 2);
            c0 += (s0 == 0u ? i0 : 0u) + (s1 == 0u ? i1 : 0u) +
                  (s2 == 0u ? i2 : 0u) + (s3 == 0u ? i3 : 0u);
            c1 += (s0 == 1u ? i0 : 0u) + (s1 == 1u ? i1 : 0u) +
                  (s2 == 1u ? i2 : 0u) + (s3 == 1u ? i3 : 0u);
            c2 += (s0 == 2u ? i0 : 0u) + (s1 == 2u ? i1 : 0u) +
                  (s2 == 2u ? i2 : 0u) + (s3 == 2u ? i3 : 0u);
        }
        // flush packed bytes into 32-bit counters
        cnt[0] += c0 & 255u; cnt[1] += (c0 >> 8) & 255u;
        cnt[2] += (c0 >> 16) & 255u; cnt[3] += c0 >> 24;
        cnt[4] += c1 & 255u; cnt[5] += (c1 >> 8) & 255u;
        cnt[6] += (c1 >> 16) & 255u; cnt[7] += c1 >> 24;
        cnt[8] += c2 & 255u; cnt[9] += (c2 >> 8) & 255u;
        c0 = c1 = c2 = 0u;
    }

    // scalar tail (n % 4), handled by block 0
    if (blockIdx.x == 0 && (int)threadIdx.x < rem) {
        long i = nvec * 4 + threadIdx.x;
        int b = bin_of(pred[i], tgt[i]);
#pragma unroll
        for (int k = 0; k < BINS; ++k) cnt[k] += (unsigned)(b == k);
    }

#pragma unroll
    for (int b = 0; b < BINS; ++b) atomicAdd(&hist[b], cnt[b]);   // ds integer atomics
    __syncthreads();
    if (threadIdx.x < BINS) atomicAdd(&gcounts[threadIdx.x], hist[threadIdx.x]);
}

// ---------------------------------------------------------------------------
// Kernel 2: per-bin weights w[b] = 1/(counts[b]*n)  (tot cancels analytically)
// ---------------------------------------------------------------------------
__global__ void k_weights(const unsigned* __restrict__ counts, float* __restrict__ w) {
    if (blockIdx.x == 0 && threadIdx.x == 0) {
        int n = 0;
        for (int b = 0; b < BINS; ++b) n += (counts[b] > 0u);
        float fn = (n > 0) ? (float)n : 1.0f;
        for (int b = 0; b < BINS; ++b)
            w[b] = (counts[b] > 0u) ? (1.0f / ((float)counts[b] * fn)) : 0.0f;
    }
}

// ---------------------------------------------------------------------------
// Kernel 3: weighted BCE partial sums. Async double-buffered LDS staging,
// per-thread fp32 accumulator -> WMMA wave sum -> fixed-order block sum.
// ---------------------------------------------------------------------------
__global__ void k_wbce(const float* __restrict__ pred, const float* __restrict__ tgt,
                       const float* __restrict__ w, float* __restrict__ partials,
                       long nvec, int rem) {
    __shared__ float4 bufP[2][TPB];
    __shared__ float4 bufT[2][TPB];
    __shared__ float  wlds[16];
    __shared__ float  wavesum[TPB / 32];

    if (threadIdx.x < 16)
        wlds[threadIdx.x] = (threadIdx.x < BINS) ? w[threadIdx.x] : 0.0f;
    __syncthreads();

    const float4* p4 = (const float4*)pred;
    const float4* t4 = (const float4*)tgt;
    long gid    = (long)blockIdx.x * blockDim.x + threadIdx.x;
    long stride = (long)gridDim.x * blockDim.x;

    float acc = 0.0f;
    int   cur = 0;

    if (gid < nvec) {                               // prologue: stage tile 0
        async_copy_b128(&bufP[0][threadIdx.x], p4 + gid);
        async_copy_b128(&bufT[0][threadIdx.x], t4 + gid);
    }
    for (long v = gid; v < nvec; v += stride) {
        long vn = v + stride;
        if (vn < nvec) {                            // stage next tile, then
            async_copy_b128(&bufP[cur ^ 1][threadIdx.x], p4 + vn);
            async_copy_b128(&bufT[cur ^ 1][threadIdx.x], t4 + vn);
            wait_async_le2();                       // current tile's 2 ops done
        } else {
            wait_async_le0();
        }
        float4 x = bufP[cur][threadIdx.x];          // ds_load_b128
        float4 t = bufT[cur][threadIdx.x];
        int   idx; float bce;
        elem_of(x.x, t.x, idx, bce); acc = fmaf(wlds[idx], bce, acc);
        elem_of(x.y, t.y, idx, bce); acc = fmaf(wlds[idx], bce, acc);
        elem_of(x.z, t.z, idx, bce); acc = fmaf(wlds[idx], bce, acc);
        elem_of(x.w, t.w, idx, bce); acc = fmaf(wlds[idx], bce, acc);
        cur ^= 1;
    }
    // scalar tail, block 0
    if (blockIdx.x == 0 && (int)threadIdx.x < rem) {
        long i = nvec * 4 + threadIdx.x;
        int idx; float bce;
        elem_of(pred[i], tgt[i], idx, bce);
        acc = fmaf(wlds[idx], bce, acc);
    }

    // exact fp32 wave sum on the matrix pipe (fully converged here)
    float wsum = wave_sum_wmma(acc);
    int lane = threadIdx.x & 31, wid = threadIdx.x >> 5;
    if (lane == 0) wavesum[wid] = wsum;
    __syncthreads();
    if (threadIdx.x == 0) {
        float s = 0.0f;
#pragma unroll
        for (int i = 0; i < TPB / 32; ++i) s += wavesum[i];   // fixed order
        partials[blockIdx.x] = s;
    }
}

// ---------------------------------------------------------------------------
// Kernel 4: deterministic fixed-mapping reduction of block partials -> scalar
// ---------------------------------------------------------------------------
__global__ void k_final(const float* __restrict__ partials, int nparts,
                        float* __restrict__ out) {
    __shared__ float wavesum[TPB / 32];
    float s = 0.0f;
    for (int i = threadIdx.x; i < nparts; i += TPB) s += partials[i];  // fixed map
    float wsum = wave_sum_wmma(s);                   // fully converged
    int lane = threadIdx.x & 31, wid = threadIdx.x >> 5;
    if (lane == 0) wavesum[wid] = wsum;
    __syncthreads();
    if (threadIdx.x == 0) {
        float tot = 0.0f;
#pragma unroll
        for (int i = 0; i < TPB / 32; ++i) tot += wavesum[i];
        out[0] = tot;   // == sum(weights*bce)/tot by construction (tot cancels)
    }
}

// ---------------------------------------------------------------------------
extern "C" void kernel_launch(void* const* d_in, const int* in_sizes, int n_in,
                              void* d_out, int out_size, void* d_ws, size_t ws_size,
                              hipStream_t stream) {
    const float* pred = (const float*)d_in[0];
    const float* tgt  = (const float*)d_in[1];
    long n    = (long)in_sizes[0];
    long nvec = n >> 2;
    int  rem  = (int)(n & 3);

    unsigned char* ws = (unsigned char*)d_ws;
    unsigned* counts   = (unsigned*)ws;          // [0,  64): u32 counts[10]
    float*    w        = (float*)(ws + 64);      // [64, 128): f32 w[16]
    float*    partials = (float*)(ws + 128);     // [128, 128+4*NBLK)

    k_zero   <<<1,    32,  0, stream>>>(counts);
    k_hist   <<<NBLK, TPB, 0, stream>>>(pred, tgt, counts, nvec, rem);
    k_weights<<<1,    32,  0, stream>>>(counts, w);
    k_wbce   <<<NBLK, TPB, 0, stream>>>(pred, tgt, w, partials, nvec, rem);
    k_final  <<<1,    TPB, 0, stream>>>(partials, NBLK, (float*)d_out);
}